// FirstOrderGCN_6038724018182
// MI455X (gfx1250) — compile-verified
//
#include <hip/hip_runtime.h>

typedef float    v2f __attribute__((ext_vector_type(2)));
typedef float    v8f __attribute__((ext_vector_type(8)));
typedef unsigned v4u __attribute__((ext_vector_type(4)));
typedef unsigned v8u __attribute__((ext_vector_type(8)));

// -------------------------------------------------------------------------
// Hardware no-return fp32 atomic add (L2 atomic ALU, STOREcnt-tracked).
// -------------------------------------------------------------------------
__device__ __forceinline__ void atomic_add_f32_hw(float* p, float v) {
    asm volatile("global_atomic_add_f32 %0, %1, off" : : "v"(p), "v"(v) : "memory");
}

// -------------------------------------------------------------------------
// Tensor Data Mover: 1-D contiguous DMA of `nelem` dwords global -> LDS.
// Builds the D# in SGPRs (group0 + group1) and issues TENSOR_LOAD_TO_LDS.
// Caller must be a single scalar-guarded wave (TDM ignores EXEC) and must
// s_wait_tensorcnt + barrier before consumers read LDS.
// -------------------------------------------------------------------------
__device__ __forceinline__ void tdm_load_1d(unsigned lds_off, const float* gsrc,
                                            unsigned nelem) {
    unsigned long long ga = (unsigned long long)(uintptr_t)gsrc;
    v4u g0;
    g0[0] = 1u;                                              // count=1, user mode
    g0[1] = lds_off;                                         // lds_addr [63:32]
    g0[2] = (unsigned)ga;                                    // global_addr lo
    g0[3] = ((unsigned)(ga >> 32) & 0x01FFFFFFu)             // global_addr hi (57b)
            | 0x80000000u;                                   // type=2 ("image")
    v8u g1;
    g1[0] = 0x00020000u;                                     // data_size=2 (4B)
    g1[1] = (nelem & 0xFFFFu) << 16;                         // tensor_dim0 lo16
    g1[2] = ((nelem >> 16) & 0xFFFFu) | (1u << 16);          // tensor_dim0 hi | tensor_dim1=1
    g1[3] = (nelem & 0xFFFFu) << 16;                         // tile_dim0 = nelem
    g1[4] = 1u;                                              // tile_dim1=1, tile_dim2=0
    g1[5] = nelem;                                           // tensor_dim0_stride lo32
    g1[6] = 0u;
    g1[7] = 0u;
    asm volatile("tensor_load_to_lds %0, %1" : : "s"(g0), "s"(g1) : "memory");
}

// LDS aperture: flat address low 32 bits == LDS byte offset (ISA 10.2).
__device__ __forceinline__ unsigned lds_offset_of(const void* p) {
    return (unsigned)(uintptr_t)p;
}

// -------------------------------------------------------------------------
// Zero a float buffer (float4 stores)
// -------------------------------------------------------------------------
__global__ void __launch_bounds__(256) zero_kernel(float4* __restrict__ p, int n4) {
    int i = blockIdx.x * blockDim.x + threadIdx.x;
    if (i < n4) p[i] = make_float4(0.f, 0.f, 0.f, 0.f);
}

// -------------------------------------------------------------------------
// SPMM: out[dst] += dis[src]*dis[dst]*x[src]   (d = 128 fp32 features)
// One wave per edge: float4/lane covers the full 512B row (coalesced).
// x (51.2MB) and out (51.2MB) are L2-resident (192MB L2), so both the
// ~16x gather reuse and the atomic RMW traffic stay off HBM.
// -------------------------------------------------------------------------
__global__ void __launch_bounds__(256) spmm_kernel(
        const float* __restrict__ x,
        const int*   __restrict__ src_idx,
        const int*   __restrict__ dst_idx,
        const float* __restrict__ dis,
        float*       __restrict__ out,
        int E) {
    int wave = blockIdx.x * (blockDim.x >> 5) + (threadIdx.x >> 5);
    int lane = threadIdx.x & 31;
    if (wave >= E) return;

    int s = src_idx[wave];
    int d = dst_idx[wave];
    float c = dis[s] * dis[d];

    const float4* xr = (const float4*)(x + (size_t)s * 128);
    float4 v = xr[lane];

    float* o = out + (size_t)d * 128 + lane * 4;
    atomic_add_f32_hw(o + 0, v.x * c);
    atomic_add_f32_hw(o + 1, v.y * c);
    atomic_add_f32_hw(o + 2, v.z * c);
    atomic_add_f32_hw(o + 3, v.w * c);
}

// -------------------------------------------------------------------------
// Fused dual GEMM: out[M,NOUT] = act(A1[M,128] @ W1[128,NOUT]
//                                  + A2[M,128] @ W2[128,NOUT])
// fp32 end-to-end via V_WMMA_F32_16X16X4_F32 (exact reference precision).
// Block = 256 threads = 8 waves; each wave owns a 16-row stripe x NOUT cols.
// Weight K-chunks staged into LDS by the Tensor Data Mover (one scalar-
// guarded wave issues tensor_load_to_lds; DMA overlaps the A-frag prefetch).
//
// Fragment layouts per ISA 7.12.2:
//   A 16x4  : lane<16 -> (A[m=lane][k], A[m=lane][k+1]);
//             lane>=16 -> (A[m=lane-16][k+2], A[m=lane-16][k+3])
//   B 4x16  : lane<16 -> (B[k][n], B[k+1][n]); lane>=16 -> (B[k+2][n], B[k+3][n])
//   C/D v8f : elem j at row (j + 8*(lane>>4)), col (lane&15) of the 16x16 tile
// -------------------------------------------------------------------------
template <int NOUT, bool RELU>
__global__ void __launch_bounds__(256) gemm_dual_kernel(
        const float* __restrict__ A1,
        const float* __restrict__ A2,
        const float* __restrict__ W1,
        const float* __restrict__ W2,
        float*       __restrict__ out,
        int M) {
    constexpr int K  = 128;
    constexpr int KC = 64;          // K-chunk staged in LDS (<= 64KB total)
    constexpr int NT = NOUT / 16;   // 16-wide output tiles per wave

    __shared__ float lw1[KC * NOUT];
    __shared__ float lw2[KC * NOUT];

    const int tid   = threadIdx.x;
    const int wave  = tid >> 5;
    const int lane  = tid & 31;
    const int lrow  = lane & 15;
    const int khalf = lane >> 4;    // 0 or 1

    int row0 = (blockIdx.x * 8 + wave) * 16;
    if (row0 + 16 > M) row0 = M - 16;   // M % 16 == 0: clamp -> duplicate stripe

    const float* a1p = A1 + (size_t)(row0 + lrow) * K + 2 * khalf;
    const float* a2p = A2 + (size_t)(row0 + lrow) * K + 2 * khalf;

    const unsigned lw1_off = lds_offset_of(lw1);
    const unsigned lw2_off = lds_offset_of(lw2);
    const bool tdm_wave = (__builtin_amdgcn_readfirstlane(wave) == 0);

    v8f acc[NT];
#pragma unroll
    for (int t = 0; t < NT; ++t) acc[t] = (v8f)(0.0f);

    for (int kc = 0; kc < K; kc += KC) {
        __syncthreads();                    // LDS from previous chunk consumed
        if (tdm_wave) {
            tdm_load_1d(lw1_off, W1 + (size_t)kc * NOUT, KC * NOUT);
            tdm_load_1d(lw2_off, W2 + (size_t)kc * NOUT, KC * NOUT);
        }
        // prefetch first A fragments of this chunk; overlaps the TDM DMA
        v2f a1 = *(const v2f*)(a1p + kc);
        v2f a2 = *(const v2f*)(a2p + kc);
        if (tdm_wave) __builtin_amdgcn_s_wait_tensorcnt(0);
        __syncthreads();                    // weights visible to all waves

        for (int k = 0; k < KC; k += 4) {
            // software-pipeline: issue next A frags before this step's WMMAs
            v2f na1 = a1, na2 = a2;
            if (k + 4 < KC) {
                na1 = *(const v2f*)(a1p + kc + k + 4);
                na2 = *(const v2f*)(a2p + kc + k + 4);
            }
            const int kb = (k + 2 * khalf) * NOUT + lrow;
#pragma unroll
            for (int t = 0; t < NT; ++t) {
                v2f b1, b2;
                b1.x = lw1[kb + t * 16];
                b1.y = lw1[kb + NOUT + t * 16];
                b2.x = lw2[kb + t * 16];
                b2.y = lw2[kb + NOUT + t * 16];
                acc[t] = __builtin_amdgcn_wmma_f32_16x16x4_f32(
                    false, a1, false, b1, (short)0, acc[t], false, false);
                acc[t] = __builtin_amdgcn_wmma_f32_16x16x4_f32(
                    false, a2, false, b2, (short)0, acc[t], false, false);
            }
            a1 = na1;
            a2 = na2;
        }
    }

    // store D tiles: elem j -> row (row0 + j + 8*khalf), col (t*16 + lrow)
    float* orow = out + (size_t)(row0 + 8 * khalf) * NOUT + lrow;
#pragma unroll
    for (int t = 0; t < NT; ++t) {
#pragma unroll
        for (int j = 0; j < 8; ++j) {
            float v = acc[t][j];
            if (RELU) v = fmaxf(v, 0.0f);
            orow[(size_t)j * NOUT + t * 16] = v;
        }
    }
}

// -------------------------------------------------------------------------
// Pipeline: zero -> spmm -> gemm(relu,128) -> zero -> spmm -> gemm(64)
// -------------------------------------------------------------------------
extern "C" void kernel_launch(void* const* d_in, const int* in_sizes, int n_in,
                              void* d_out, int out_size, void* d_ws, size_t ws_size,
                              hipStream_t stream) {
    const float* x     = (const float*)d_in[0];
    const int*   edges = (const int*)d_in[1];   // [2,E] int32 (JAX x64 disabled)
    const float* dis   = (const float*)d_in[2];
    // d_in[3] = num_nodes (unused; derived from in_sizes)
    const float* W0_1  = (const float*)d_in[4];
    const float* W1_1  = (const float*)d_in[5];
    const float* W0_2  = (const float*)d_in[6];
    const float* W1_2  = (const float*)d_in[7];
    float* out = (float*)d_out;

    const int N = in_sizes[0] / 128;
    const int E = in_sizes[1] / 2;
    const int* src = edges;
    const int* dst = edges + E;

    float* Abuf = (float*)d_ws;                  // N*128 floats (A_x / A_out1)
    float* out1 = Abuf + (size_t)N * 128;        // N*128 floats

    const int n4      = N * 128 / 4;
    const int zblocks = (n4 + 255) / 256;
    const int eblocks = (E + 7) / 8;             // 8 waves (edges) per block
    const int gblocks = (N / 16 + 7) / 8;        // 8 row-stripes per block

    // Layer 1
    zero_kernel<<<zblocks, 256, 0, stream>>>((float4*)Abuf, n4);
    spmm_kernel<<<eblocks, 256, 0, stream>>>(x, src, dst, dis, Abuf, E);
    gemm_dual_kernel<128, true><<<gblocks, 256, 0, stream>>>(
        x, Abuf, W0_1, W1_1, out1, N);

    // Layer 2
    zero_kernel<<<zblocks, 256, 0, stream>>>((float4*)Abuf, n4);
    spmm_kernel<<<eblocks, 256, 0, stream>>>(out1, src, dst, dis, Abuf, E);
    gemm_dual_kernel<64, false><<<gblocks, 256, 0, stream>>>(
        out1, Abuf, W0_2, W1_2, out, N);
}